// WeightedMIRLLoss_29583734735545
// MI455X (gfx1250) — compile-verified
//
#include <hip/hip_runtime.h>

#define MARGIN_V 0.2f

typedef __attribute__((ext_vector_type(2))) float v2f;
typedef __attribute__((ext_vector_type(4))) float v4f;
typedef __attribute__((ext_vector_type(8))) float v8f;

#define ROWS 16
#define TILE_W 128
#define TILE_STRIDE 132  // 132%64=4 -> A-fragment b64 reads hit banks 4L+kc..+3: conflict-free

__global__ __launch_bounds__(256)
void mirl_rows_kernel(const float* __restrict__ scores,
                      const float* __restrict__ wts,
                      const int* __restrict__ gt,
                      const unsigned char* __restrict__ valid,
                      float* __restrict__ row_loss,
                      float* __restrict__ row_ok,
                      int N)
{
    __shared__ float tileL[ROWS * TILE_STRIDE];   // masked weighted losses
    __shared__ float tileM[ROWS * TILE_STRIDE];   // 0/1 negative mask
    __shared__ float pos_s[ROWS];
    __shared__ int   gtc_s[ROWS];
    __shared__ int   ok_s[ROWS];
    __shared__ float partL[8][ROWS];
    __shared__ float partC[8][ROWS];

    const int tid  = threadIdx.x;
    const int wave = tid >> 5;
    const int lane = tid & 31;
    const int row0 = blockIdx.x * ROWS;

    // Per-row setup: pos score, clipped gt index, sample_ok
    if (tid < ROWS) {
        const int r = row0 + tid;
        const int g = gt[r];
        const int in_range = (g >= 0) && (g < N);
        int gc = g;
        if (gc < 0) gc = 0;
        if (gc > N - 1) gc = N - 1;
        gtc_s[tid] = gc;
        pos_s[tid] = scores[(size_t)r * N + gc];
        ok_s[tid]  = in_range && (valid[(size_t)r * N + gc] != 0);
    }
    __syncthreads();

    // Each thread owns one row and 4 consecutive columns per 64-col half.
    const int row   = tid >> 4;        // 0..15
    const int cbase = (tid & 15) * 4;  // 0..60
    const float pos_r = pos_s[row];    // hoisted: no LDS reads in hot loop
    const int   gtc_r = gtc_s[row];

    v8f accL = {};   // per-row sum of masked losses (col 0 of D is the row sum)
    v8f accC = {};   // per-row negative count
    v2f ones; ones.x = 1.0f; ones.y = 1.0f;   // B = 4x16 all-ones

    const size_t rbase = (size_t)(row0 + row) * N;
    const int nchunk = N / TILE_W;

    for (int chunk = 0; chunk < nchunk; ++chunk) {
        const int c0 = chunk * TILE_W;

        // Streaming phase: b128 loads, 256B contiguous per 16-lane row segment
        #pragma unroll
        for (int half = 0; half < 2; ++half) {
            const int ccol = half * 64 + cbase;
            const int gcol = c0 + ccol;
            const size_t gidx = rbase + gcol;
            const v4f s = *(const v4f*)(scores + gidx);
            const v4f w = *(const v4f*)(wts + gidx);
            const unsigned int vm = *(const unsigned int*)(valid + gidx);

            v4f l, m;
            #pragma unroll
            for (int j = 0; j < 4; ++j) {
                const int ok = (int)(((vm >> (8 * j)) & 0xFFu) != 0u) &
                               (int)((gcol + j) != gtc_r);
                const float loss = fmaxf(MARGIN_V - (pos_r - s[j]), 0.0f) * w[j];
                l[j] = ok ? loss : 0.0f;
                m[j] = ok ? 1.0f : 0.0f;
            }
            *(v4f*)(tileL + row * TILE_STRIDE + ccol) = l;
            *(v4f*)(tileM + row * TILE_STRIDE + ccol) = m;

            if (chunk + 1 < nchunk) {
                __builtin_prefetch(scores + gidx + TILE_W, 0, 1);  // global_prefetch_b8
                __builtin_prefetch(wts + gidx + TILE_W, 0, 1);
            }
        }
        __syncthreads();

        // WMMA row-reduction: wave w owns cols [w*16, w*16+16) of this chunk.
        // A layout (16x4 f32): lane<16 -> M=lane, K=0..1; lane>=16 -> M=lane-16, K=2..3
        const int arow = lane & 15;
        const int koff = (lane >> 4) << 1;
        #pragma unroll
        for (int t = 0; t < 4; ++t) {
            const int kc   = wave * 16 + t * 4;
            const int base = arow * TILE_STRIDE + kc + koff;
            v2f aL; aL.x = tileL[base]; aL.y = tileL[base + 1];
            v2f aC; aC.x = tileM[base]; aC.y = tileM[base + 1];
            accL = __builtin_amdgcn_wmma_f32_16x16x4_f32(
                       false, aL, false, ones, (short)0, accL, false, false);
            accC = __builtin_amdgcn_wmma_f32_16x16x4_f32(
                       false, aC, false, ones, (short)0, accC, false, false);
        }
        __syncthreads();
    }

    // D layout: VGPR i, lane 0 -> D[M=i][N=0]; lane 16 -> D[M=i+8][N=0]
    if (lane == 0) {
        #pragma unroll
        for (int i = 0; i < 8; ++i) { partL[wave][i] = accL[i]; partC[wave][i] = accC[i]; }
    } else if (lane == 16) {
        #pragma unroll
        for (int i = 0; i < 8; ++i) { partL[wave][8 + i] = accL[i]; partC[wave][8 + i] = accC[i]; }
    }
    __syncthreads();

    if (tid < ROWS) {
        float sl = 0.0f, sc = 0.0f;
        #pragma unroll
        for (int w = 0; w < 8; ++w) { sl += partL[w][tid]; sc += partC[w][tid]; }
        float rl = 0.0f;
        if (ok_s[tid] && sc > 0.0f) rl = sl / sc;   // == sl / max(count,1) when count>0
        row_loss[row0 + tid] = rl;
        row_ok[row0 + tid]   = ok_s[tid] ? 1.0f : 0.0f;
    }
}

__global__ __launch_bounds__(256)
void mirl_final_kernel(const float* __restrict__ row_loss,
                       const float* __restrict__ row_ok,
                       float* __restrict__ out, int B)
{
    __shared__ float sl[256];
    __shared__ float sc[256];
    float al = 0.0f, ac = 0.0f;
    for (int i = threadIdx.x; i < B; i += 256) { al += row_loss[i]; ac += row_ok[i]; }
    sl[threadIdx.x] = al; sc[threadIdx.x] = ac;
    __syncthreads();
    for (int s = 128; s > 0; s >>= 1) {
        if ((int)threadIdx.x < s) {
            sl[threadIdx.x] += sl[threadIdx.x + s];
            sc[threadIdx.x] += sc[threadIdx.x + s];
        }
        __syncthreads();
    }
    if (threadIdx.x == 0) {
        const float sum = sl[0];
        const float vs  = sc[0];
        const float ranking = (vs > 0.0f) ? (sum / fmaxf(vs, 1.0f)) : sum;
        out[0] = ranking;   // total = ranking + LAMBDA_REJECT * 0
        out[1] = ranking;   // ranking
        out[2] = 0.0f;      // rejection
    }
}

extern "C" void kernel_launch(void* const* d_in, const int* in_sizes, int n_in,
                              void* d_out, int out_size, void* d_ws, size_t ws_size,
                              hipStream_t stream) {
    const float*         scores = (const float*)d_in[0];
    const float*         wts    = (const float*)d_in[1];
    const int*           gt     = (const int*)d_in[2];
    const unsigned char* valid  = (const unsigned char*)d_in[3];

    const int B = in_sizes[2];
    const int N = in_sizes[0] / B;

    float* row_loss = (float*)d_ws;       // B floats
    float* row_ok   = row_loss + B;       // B floats

    mirl_rows_kernel<<<B / ROWS, 256, 0, stream>>>(scores, wts, gt, valid,
                                                   row_loss, row_ok, N);
    mirl_final_kernel<<<1, 256, 0, stream>>>(row_loss, row_ok, (float*)d_out, B);
}